// ROIWarp_85358180041455
// MI455X (gfx1250) — compile-verified
//
#include <hip/hip_runtime.h>
#include <hip/hip_bf16.h>
#include <cstdint>

#define OUT_HW 56
#define HF 128
#define WF 128

typedef __attribute__((ext_vector_type(2))) float v2f;
typedef __attribute__((ext_vector_type(8))) float v8f;

#if defined(__HIP_DEVICE_COMPILE__) && \
    __has_builtin(__builtin_amdgcn_global_load_async_to_lds_b32) && \
    __has_builtin(__builtin_amdgcn_s_wait_asynccnt)
#define USE_ASYNC_LDS 1
typedef __attribute__((address_space(1))) int gi_t;   // global int
typedef __attribute__((address_space(3))) int li_t;   // LDS int
#else
#define USE_ASYNC_LDS 0
#endif

// One workgroup (256 threads = 8 wave32) per ROI.
// out[56,56] = Wy(56xbh) * crop(bhxbw) * WxT(bwx56), all f32, via
// V_WMMA_F32_16X16X4_F32 tiles padded to 64x36x48 / 64x36x64.
// Crop staging is double-buffered through GLOBAL_LOAD_ASYNC_TO_LDS (ASYNCcnt),
// overlapping the async engine with the WMMA pipeline.
__launch_bounds__(256, 2)
__global__ void ROIWarp_85358180041455_kernel(const float* __restrict__ fm,
                                              const int* __restrict__ boxes,
                                              float* __restrict__ out,
                                              int M, int C)
{
    __shared__ float sWy[64][37];      // Wy[oy][ysrc]; odd stride -> conflict-free A reads
    __shared__ float sWxT[36][64];     // WxT[xsrc][ox]
    __shared__ float sCrop[2][36][48]; // double-buffered zero-padded crop
    __shared__ float sT[64][49];       // T = Wy * crop; odd stride

    const int b    = blockIdx.x;       // roi index over (S*N, M)
    const int sn   = b / M;            // image index
    const int tid  = threadIdx.x;
    const int lane = tid & 31;
    const int wave = tid >> 5;

    const int x1 = boxes[b * 4 + 0];
    const int y1 = boxes[b * 4 + 1];
    const int bw = boxes[b * 4 + 2] - x1;   // 8..33
    const int bh = boxes[b * 4 + 3] - y1;   // 8..33

    const long fm_img  = (long)sn * C * (HF * WF);
    const long out_roi = (long)b * C * (OUT_HW * OUT_HW);

    // Issue the (EXEC-masked) crop copy for channel c into buffer buf.
    auto crop_load = [&](int buf, int c) {
        const float* src = fm + fm_img + (long)c * (HF * WF);
        float* cb = &sCrop[buf][0][0];
        for (int i = tid; i < 36 * 48; i += 256) {
            const int y = i / 48, x = i - y * 48;
            if (y < bh && x < bw) {
                const float* g = src + (long)(y1 + y) * WF + (x1 + x);
#if USE_ASYNC_LDS
                __builtin_amdgcn_global_load_async_to_lds_b32(
                    (gi_t*)(uintptr_t)g, (li_t*)(uintptr_t)(cb + i), 0, 0);
#else
                cb[i] = *g;
#endif
            }
        }
    };

    // ---- zero weight matrices and both crop buffers (padding halo stays 0) ----
    {
        float* p0 = &sWy[0][0];
        for (int i = tid; i < 64 * 37; i += 256) p0[i] = 0.0f;
        float* p1 = &sWxT[0][0];
        for (int i = tid; i < 36 * 64; i += 256) p1[i] = 0.0f;
        float* p2 = &sCrop[0][0][0];
        for (int i = tid; i < 2 * 36 * 48; i += 256) p2[i] = 0.0f;
    }
    __syncthreads();

    // ---- build interpolation matrices (PyTorch bilinear, align_corners=False) ----
    if (tid < OUT_HW) {                           // y axis: thread owns output row oy
        const int oy = tid;
        const float ext = (float)bh;
        float r = ((float)oy + 0.5f) * ext / (float)OUT_HW - 0.5f;
        float hi = ext - 1.0f; if (hi < 0.0f) hi = 0.0f;
        r = fminf(fmaxf(r, 0.0f), hi);
        int i0 = (int)floorf(r);
        int mx = bh - 1; if (mx < 0) mx = 0;
        int i1 = i0 + 1; if (i1 > mx) i1 = mx;
        float w = r - (float)i0;
        sWy[oy][i0] += 1.0f - w;
        sWy[oy][i1] += w;
    } else if (tid >= 64 && tid < 64 + OUT_HW) {  // x axis: thread owns output col ox
        const int ox = tid - 64;
        const float ext = (float)bw;
        float r = ((float)ox + 0.5f) * ext / (float)OUT_HW - 0.5f;
        float hi = ext - 1.0f; if (hi < 0.0f) hi = 0.0f;
        r = fminf(fmaxf(r, 0.0f), hi);
        int i0 = (int)floorf(r);
        int mx = bw - 1; if (mx < 0) mx = 0;
        int i1 = i0 + 1; if (i1 > mx) i1 = mx;
        float w = r - (float)i0;
        sWxT[i0][ox] += 1.0f - w;
        sWxT[i1][ox] += w;
    }

    // Prologue: start streaming channel 0's crop.
    crop_load(0, 0);

    const int row = lane & 15;          // within-tile row (A) / col (B)
    const int ks  = (lane >> 4) << 1;   // K sub-offset: lanes 0-15 -> K=0,1; 16-31 -> K=2,3
    const int hiM = (lane >> 4) << 3;   // D-layout row offset for upper half-wave

    for (int c = 0; c < C; ++c) {
        const int buf = c & 1;

#if USE_ASYNC_LDS
        __builtin_amdgcn_s_wait_asynccnt(0);   // my async loads for channel c landed
#endif
        __syncthreads();                       // everyone's crop (and weights) visible

        // Kick off the async stream for the next channel into the other buffer;
        // it overlaps with both WMMA stages below.
        if (c + 1 < C)
            crop_load(buf ^ 1, c + 1);

        // ---- stage 1: T(64x48) = Wy(64x36) @ crop(36x48) ----
        for (int t = wave; t < 12; t += 8) {        // 4 row-tiles x 3 col-tiles
            const int oy0 = (t / 3) * 16;
            const int x0  = (t % 3) * 16;
            v8f acc = {};
            #pragma unroll
            for (int kk = 0; kk < 9; ++kk) {
                const int k = kk * 4 + ks;
                v2f a;  a.x = sWy[oy0 + row][k];        a.y = sWy[oy0 + row][k + 1];
                v2f bb; bb.x = sCrop[buf][k][x0 + row]; bb.y = sCrop[buf][k + 1][x0 + row];
                acc = __builtin_amdgcn_wmma_f32_16x16x4_f32(
                        false, a, false, bb, (short)0, acc, false, false);
            }
            #pragma unroll
            for (int v = 0; v < 8; ++v)
                sT[oy0 + hiM + v][x0 + row] = acc[v];
        }
        __syncthreads();

        // ---- stage 2: out(64x64) = T(64x36) @ WxT(36x64), clipped to 56x56 ----
        float* dst = out + out_roi + (long)c * (OUT_HW * OUT_HW);
        for (int t = wave; t < 16; t += 8) {        // 4 x 4 tiles
            const int oy0 = (t >> 2) * 16;
            const int ox0 = (t & 3) * 16;
            v8f acc = {};
            #pragma unroll
            for (int kk = 0; kk < 9; ++kk) {
                const int k = kk * 4 + ks;
                v2f a;  a.x = sT[oy0 + row][k];      a.y = sT[oy0 + row][k + 1];
                v2f bb; bb.x = sWxT[k][ox0 + row];   bb.y = sWxT[k + 1][ox0 + row];
                acc = __builtin_amdgcn_wmma_f32_16x16x4_f32(
                        false, a, false, bb, (short)0, acc, false, false);
            }
            const int ox = ox0 + row;
            if (ox < OUT_HW) {
                #pragma unroll
                for (int v = 0; v < 8; ++v) {
                    const int oy = oy0 + hiM + v;
                    if (oy < OUT_HW)
                        dst[oy * OUT_HW + ox] = acc[v];
                }
            }
        }
        __syncthreads();   // protect sT (and crop buffer reuse) before next channel
    }
}

extern "C" void kernel_launch(void* const* d_in, const int* in_sizes, int n_in,
                              void* d_out, int out_size, void* d_ws, size_t ws_size,
                              hipStream_t stream) {
    const float* fm    = (const float*)d_in[0];   // [S,N,C,128,128] f32
    const int*   boxes = (const int*)d_in[1];     // [S,N,M,4] i32
    float*       out   = (float*)d_out;           // [S,N*M,C,56,56] f32

    const int n_sn = in_sizes[2];                       // S*N (has_box_indx is S*N)
    const int M    = in_sizes[1] / (4 * n_sn);          // boxes per image
    const int C    = in_sizes[0] / (n_sn * HF * WF);    // channels
    const int nroi = n_sn * M;                          // one block per ROI

    ROIWarp_85358180041455_kernel<<<nroi, 256, 0, stream>>>(fm, boxes, out, M, C);
}